// AttnWeightedKMedoidsPool_66082366816344
// MI455X (gfx1250) — compile-verified
//
#include <hip/hip_runtime.h>
#include <cstdint>

#define B_    8
#define F_    256
#define S_    1024
#define KC    16
#define NITER 5
#define CF    8           // feature chunk for pairwise-L1 LDS tiling
#define FMAX  3.402823466e+38f

typedef __attribute__((ext_vector_type(2))) float v2f;
typedef __attribute__((ext_vector_type(8))) float v8f;

// -------- wm[b,j] = mean_i w[b,i,j] ----------------------------------------
// w is 32 MB read exactly once: NT hint keeps it from evicting L2-resident d.
__global__ void wm_kernel(const float* __restrict__ w, float* __restrict__ wm) {
    int idx = blockIdx.x * blockDim.x + threadIdx.x;      // 0..B*S-1
    int b = idx / S_, col = idx % S_;
    const float* wp = w + (size_t)b * S_ * S_ + col;
    float s = 0.f;
    for (int i = 0; i < S_; ++i) s += __builtin_nontemporal_load(wp + (size_t)i * S_);
    wm[idx] = s * (1.0f / S_);
}

// -------- d[b,i,j] = sum_f |x[b,f,i] - x[b,f,j]| ---------------------------
__global__ void pdist_kernel(const float* __restrict__ x, float* __restrict__ d) {
    int b  = blockIdx.z;
    int i0 = blockIdx.y * 32;
    int j0 = blockIdx.x * 32;
    __shared__ float xi[CF][32];
    __shared__ float xj[CF][32];
    int t = threadIdx.x;              // 256 threads
    int tj = t & 31, trow = t >> 5;   // trow 0..7
    float acc[4] = {0.f, 0.f, 0.f, 0.f};
    const float* xb = x + (size_t)b * F_ * S_;
    for (int f0 = 0; f0 < F_; f0 += CF) {
        xi[trow][tj] = xb[(size_t)(f0 + trow) * S_ + i0 + tj];
        xj[trow][tj] = xb[(size_t)(f0 + trow) * S_ + j0 + tj];
        if (f0 + CF < F_) {  // stream next chunk toward L2 (global_prefetch_b8)
            __builtin_prefetch(xb + (size_t)(f0 + CF + trow) * S_ + i0 + tj, 0, 1);
            __builtin_prefetch(xb + (size_t)(f0 + CF + trow) * S_ + j0 + tj, 0, 1);
        }
        __syncthreads();
#pragma unroll
        for (int ff = 0; ff < CF; ++ff) {
            float vj = xj[ff][tj];
#pragma unroll
            for (int r = 0; r < 4; ++r)
                acc[r] += fabsf(xi[ff][trow + 8 * r] - vj);
        }
        __syncthreads();
    }
    float* dp = d + (size_t)b * S_ * S_;
#pragma unroll
    for (int r = 0; r < 4; ++r)
        dp[(size_t)(i0 + trow + 8 * r) * S_ + j0 + tj] = acc[r];
}

// -------- initial medoids: indices of K largest wm[b,:] (stable ties) ------
__global__ void topk_kernel(const float* __restrict__ wm, int* __restrict__ ctr) {
    int b = blockIdx.x;
    __shared__ float sw[S_];
    __shared__ float rv[256];
    __shared__ int   ri[256];
    int t = threadIdx.x;
    for (int j = t; j < S_; j += 256) sw[j] = wm[b * S_ + j];
    __syncthreads();
    for (int kk = 0; kk < KC; ++kk) {
        float bv = -FMAX; int bi = S_;
        for (int j = t; j < S_; j += 256) {
            float v = sw[j];
            if (v > bv) { bv = v; bi = j; }   // ascending scan -> smallest idx on tie
        }
        rv[t] = bv; ri[t] = bi;
        __syncthreads();
        for (int off = 128; off > 0; off >>= 1) {
            if (t < off) {
                float v2 = rv[t + off]; int i2 = ri[t + off];
                if (v2 > rv[t] || (v2 == rv[t] && i2 < ri[t])) { rv[t] = v2; ri[t] = i2; }
            }
            __syncthreads();
        }
        if (t == 0) { ctr[b * KC + kk] = ri[0]; sw[ri[0]] = -FMAX; }
        __syncthreads();
    }
}

// -------- assign[b,s] = argmin_k d[b,s,ctr[b,k]] ---------------------------
__global__ void assign_kernel(const float* __restrict__ d, const int* __restrict__ ctr,
                              int* __restrict__ assign) {
    int idx = blockIdx.x * blockDim.x + threadIdx.x;      // 0..B*S-1
    int b = idx / S_, s = idx % S_;
    const float* drow = d + (size_t)b * S_ * S_ + (size_t)s * S_;
    float best = FMAX; int bk = 0;
    for (int k = 0; k < KC; ++k) {
        float v = drow[ctr[b * KC + k]];
        if (v < best) { best = v; bk = k; }   // strict < -> first min like jnp.argmin
    }
    assign[idx] = bk;
}

// -------- bw[b][j/2][n][j&1] = wm[b,j] * [assign[b,j]==n] ------------------
// Pair-interleaved layout: lane's two B elements per WMMA step are one b64.
// Unconditional load + v_cndmask select: no divergent loads in any hot path.
__global__ void bprep_kernel(const float* __restrict__ wm, const int* __restrict__ assign,
                             float* __restrict__ bw) {
    int idx = blockIdx.x * blockDim.x + threadIdx.x;      // 0..B*S*KC-1
    int n  = idx & (KC - 1);
    int j  = (idx >> 4) & (S_ - 1);
    int b  = idx >> 14;
    float v = wm[b * S_ + j];
    int   a = assign[b * S_ + j];
    size_t pos = (((size_t)b * (S_ / 2) + (j >> 1)) * KC + n) * 2 + (j & 1);
    bw[pos] = (a == n) ? v : 0.0f;
}

// -------- cost[b,i,k] = sum_j d[b,i,j] * bw-masked wm ----------------------
// One wave per 16-row strip; K=16 is one N tile; S/4 chained f32 WMMAs.
// Per step: 1x global_load_b64 (A) + 1x global_load_b64 (B) + 1x v_wmma.
__global__ void __launch_bounds__(32) cost_wmma_kernel(
        const float* __restrict__ d, const float* __restrict__ bw,
        float* __restrict__ cost) {
    int b    = blockIdx.y;
    int i0   = blockIdx.x * 16;
    int lane = threadIdx.x;
    int half = lane >> 4;       // 0: K pair {0,1}; 1: K pair {2,3}
    int l    = lane & 15;       // A row m and B col n
    int k0   = half * 2;
    // A: row i0+l of d, starting at column k0 (8-byte aligned: k0 even).
    const float* ap = d  + (size_t)b * S_ * S_ + (size_t)(i0 + l) * S_ + k0;
    // B: pair-row (j+k0)/2, col l, both parity elements (8-byte aligned).
    const float* bp = bw + (size_t)b * (S_ / 2) * (KC * 2) + (size_t)half * (KC * 2) + l * 2;
    v8f acc = {};
    for (int j = 0; j < S_; j += 16) {
#pragma unroll
        for (int u = 0; u < 4; ++u) {
            int jj = j + 4 * u;
            v2f av = *(const v2f*)(ap + jj);             // d[i0+l, jj+k0 .. jj+k0+1]
            v2f bv = *(const v2f*)(bp + (size_t)jj * (KC / 2) * 2 * 2); // pair-row jj + half
            acc = __builtin_amdgcn_wmma_f32_16x16x4_f32(
                      false, av, false, bv, (short)0, acc, false, false);
        }
    }
    float* cp = cost + ((size_t)b * S_ + i0) * KC;
#pragma unroll
    for (int r = 0; r < 8; ++r) {
        int row = r + half * 8;                   // C/D layout: VGPR r -> M = r + 8*half
        cp[row * KC + l] = acc[r];
    }
}

// -------- ctr[b,k] = argmin over members i of cost[b,i,k] ------------------
__global__ void newctr_kernel(const float* __restrict__ cost, const int* __restrict__ assign,
                              int* __restrict__ ctr) {
    int b = blockIdx.x / KC, k = blockIdx.x % KC;
    __shared__ float rv[256];
    __shared__ int   ri[256];
    int t = threadIdx.x;
    float best = FMAX; int bi = 0x7FFFFFFF;
    for (int i = t; i < S_; i += 256) {
        if (assign[b * S_ + i] == k) {
            float v = cost[((size_t)b * S_ + i) * KC + k];
            if (v < best || (v == best && i < bi)) { best = v; bi = i; }
        }
    }
    rv[t] = best; ri[t] = bi;
    __syncthreads();
    for (int off = 128; off > 0; off >>= 1) {
        if (t < off) {
            float v2 = rv[t + off]; int i2 = ri[t + off];
            if (v2 < rv[t] || (v2 == rv[t] && i2 < ri[t])) { rv[t] = v2; ri[t] = i2; }
        }
        __syncthreads();
    }
    if (t == 0) ctr[b * KC + k] = (ri[0] == 0x7FFFFFFF) ? 0 : ri[0];  // empty cluster -> 0
}

// -------- out[b,f,k] = x[b,f,ctr[b,k]] -------------------------------------
__global__ void gather_kernel(const float* __restrict__ x, const int* __restrict__ ctr,
                              float* __restrict__ out) {
    int idx = blockIdx.x * blockDim.x + threadIdx.x;      // 0..B*F*K-1
    int b = idx / (F_ * KC);
    int f = (idx / KC) % F_;
    int k = idx % KC;
    out[idx] = x[(size_t)b * F_ * S_ + (size_t)f * S_ + ctr[b * KC + k]];
}

extern "C" void kernel_launch(void* const* d_in, const int* in_sizes, int n_in,
                              void* d_out, int out_size, void* d_ws, size_t ws_size,
                              hipStream_t stream) {
    (void)in_sizes; (void)n_in; (void)out_size; (void)ws_size;
    const float* x = (const float*)d_in[0];   // [B,F,S]
    const float* w = (const float*)d_in[1];   // [B,S,S]
    float* out = (float*)d_out;               // [B,F,K]

    char* ws = (char*)d_ws;
    float* d_d    = (float*)ws; ws += (size_t)B_ * S_ * S_ * sizeof(float);   // 32 MB (L2-resident)
    float* d_wm   = (float*)ws; ws += (size_t)B_ * S_ * sizeof(float);
    float* d_cost = (float*)ws; ws += (size_t)B_ * S_ * KC * sizeof(float);
    float* d_bw   = (float*)ws; ws += (size_t)B_ * S_ * KC * sizeof(float);   // 512 KB masked-B operand
    int*   d_ctr  = (int*)ws;   ws += (size_t)B_ * KC * sizeof(int);
    int*   d_asn  = (int*)ws;   ws += (size_t)B_ * S_ * sizeof(int);

    wm_kernel<<<B_ * S_ / 256, 256, 0, stream>>>(w, d_wm);
    dim3 pg(S_ / 32, S_ / 32, B_);
    pdist_kernel<<<pg, 256, 0, stream>>>(x, d_d);
    topk_kernel<<<B_, 256, 0, stream>>>(d_wm, d_ctr);

    for (int it = 0; it < NITER; ++it) {
        assign_kernel<<<B_ * S_ / 256, 256, 0, stream>>>(d_d, d_ctr, d_asn);
        bprep_kernel<<<B_ * S_ * KC / 256, 256, 0, stream>>>(d_wm, d_asn, d_bw);
        dim3 cg(S_ / 16, B_);
        cost_wmma_kernel<<<cg, 32, 0, stream>>>(d_d, d_bw, d_cost);
        newctr_kernel<<<B_ * KC, 256, 0, stream>>>(d_cost, d_asn, d_ctr);
    }

    gather_kernel<<<B_ * F_ * KC / 256, 256, 0, stream>>>(x, d_ctr, out);
}